// OffroadLossPolygons_24670292148303
// MI455X (gfx1250) — compile-verified
//
#include <hip/hip_runtime.h>
#include <math.h>

// Problem dims (fixed by the reference)
#define NB 16      // batch
#define NM 16      // polygons per batch
#define NV 200     // vertices per polygon
#define NSEG 199   // segments per polygon
#define NA 10      // agents
#define NP 30      // points per agent
#define ROWF 404   // padded floats per polygon row in LDS (400 + 4 pad -> bank spread)
#define BLOCK 480  // 15 wave32 waves; one thread per (m, p) tuple: 16*30 = 480

#define EPSF 1e-6f
#define THRESH 0.5f

// Parameter types for __builtin_amdgcn_global_load_async_to_lds_b128, per the
// compiler diagnostic: (v4i addrspace(1)*, v4i addrspace(3)*, imm offset, imm cpol)
typedef int v4i __attribute__((vector_size(4 * sizeof(int))));
typedef __attribute__((address_space(1))) v4i global_v4i;
typedef __attribute__((address_space(3))) v4i lds_v4i;

__global__ __launch_bounds__(BLOCK) void offroad_kernel(
    const float* __restrict__ points,
    const float* __restrict__ polys,
    float* __restrict__ out)
{
    __shared__ float    sPoly[NM * ROWF];   // 25,856 B: all 16 polygons of batch b
    __shared__ unsigned sDistBits[NP];      // per-point min dist^2 (as ordered uint bits)
    __shared__ unsigned sCross[NP];         // per-point crossing count (parity matters)
    __shared__ float    sVal[NP];           // per-point final relu value

    const int tid = threadIdx.x;
    const int b   = blockIdx.x;             // 0..15
    const int a   = blockIdx.y;             // 0..9

    const float* polyB = polys + (size_t)b * (NM * NV * 2);

    if (tid < NP) { sDistBits[tid] = 0x7F7FFFFFu; sCross[tid] = 0u; }

    // ---- Stage batch-b polygons into LDS: 16 rows x 1600 B = 100 x b128 per row ----
    for (int i = tid; i < NM * 100; i += BLOCK) {
        const int m = i / 100;
        const int c = i % 100;
        const float* g = polyB + m * (NV * 2) + c * 4;
        float*       l = &sPoly[m * ROWF + c * 4];
#if __has_builtin(__builtin_amdgcn_global_load_async_to_lds_b128)
        __builtin_amdgcn_global_load_async_to_lds_b128(
            (global_v4i*)g, (lds_v4i*)l, /*offset=*/0, /*cpol=*/0);
#else
        l[0] = g[0]; l[1] = g[1]; l[2] = g[2]; l[3] = g[3];
#endif
    }
#if __has_builtin(__builtin_amdgcn_s_wait_asynccnt)
    __builtin_amdgcn_s_wait_asynccnt(0);
#else
    asm volatile("s_wait_asynccnt 0" ::: "memory");
#endif
    __syncthreads();

    // ---- One thread per (m, p): 199-segment loop entirely out of LDS ----
    const int p = tid % NP;   // 0..29
    const int m = tid / NP;   // 0..15

    const float2 pt = ((const float2*)points)[(b * NA + a) * NP + p];
    const float px = pt.x, py = pt.y;

    const float* seg = &sPoly[m * ROWF];
    float mind2 = 3.402823466e38f;
    unsigned cross = 0;

    float sx = seg[0], sy = seg[1];
    for (int s = 0; s < NSEG; ++s) {
        const float ex = seg[2 * s + 2];
        const float ey = seg[2 * s + 3];
        const float evx = ex - sx, evy = ey - sy;
        const float vx  = px - sx, vy  = py - sy;

        // distance to segment (reference order: clip(dot/(esq+eps), 0, 1))
        const float dot = vx * evx + vy * evy;
        const float esq = evx * evx + evy * evy;
        float proj = dot / (esq + EPSF);
        proj = fminf(fmaxf(proj, 0.0f), 1.0f);
        const float cx = evx * proj + sx;
        const float cy = evy * proj + sy;
        const float dx = px - cx, dy = py - cy;
        const float d2 = fmaxf(dx * dx + dy * dy, EPSF);
        mind2 = fminf(mind2, d2);

        // ray crossing test, replicated exactly (incl. the double-eps quirk)
        const bool cond_y = ((sy <= py) && (py < ey)) || ((ey <= py) && (py < sy));
        const float slope = (ey - sy) / (ex - sx + EPSF);
        const float ix    = sx + (py - sy) / (slope + EPSF);
        cross += (cond_y && (ix > px)) ? 1u : 0u;

        sx = ex; sy = ey;
    }

    // ---- Reduce over m via LDS atomics (ds_min_u32 / ds_add_u32) ----
    // mind2 >= EPS > 0, so uint-bit ordering == float ordering.
    atomicMin(&sDistBits[p], __float_as_uint(mind2));
    atomicAdd(&sCross[p], cross);
    __syncthreads();

    // ---- Per-point finalize ----
    if (tid < NP) {
        const float d = sqrtf(__uint_as_float(sDistBits[tid]));
        float v = (sCross[tid] & 1u) ? -d : d;
        v += THRESH;
        sVal[tid] = v > 0.0f ? v : 0.0f;
    }
    __syncthreads();

    // ---- Sum over the 30 points with a wave32 shuffle tree ----
    if (tid < 32) {
        float v = (tid < NP) ? sVal[tid] : 0.0f;
        for (int off = 16; off > 0; off >>= 1)
            v += __shfl_xor(v, off, 32);
        if (tid == 0) out[b * NA + a] = v;
    }
}

extern "C" void kernel_launch(void* const* d_in, const int* in_sizes, int n_in,
                              void* d_out, int out_size, void* d_ws, size_t ws_size,
                              hipStream_t stream) {
    (void)in_sizes; (void)n_in; (void)out_size; (void)d_ws; (void)ws_size;
    const float* points = (const float*)d_in[0];   // (16,10,30,2) f32
    const float* polys  = (const float*)d_in[1];   // (16,16,200,2) f32
    float* out          = (float*)d_out;           // (16,10) f32

    dim3 grid(NB, NA);
    offroad_kernel<<<grid, BLOCK, 0, stream>>>(points, polys, out);
}